// GCN_51213190037828
// MI455X (gfx1250) — compile-verified
//
#include <hip/hip_runtime.h>

typedef float v2f __attribute__((ext_vector_type(2)));
typedef float v4f __attribute__((ext_vector_type(4)));
typedef float v8f __attribute__((ext_vector_type(8)));
typedef int   v4i __attribute__((vector_size(16)));              // gcc-style, matches builtin proto
typedef __attribute__((address_space(1))) v4i gv4i;              // global
typedef __attribute__((address_space(3))) v4i lv4i;              // LDS

#define B_ 8
#define N_ 2048
#define D_ 256
#define LDA 36    // A tile row stride (dwords): 144B rows -> 16B aligned, conflict-free frag reads
#define LDBP 544  // paired-B row stride (dwords): 544%64==32 -> hi/lo lane halves hit disjoint banks

// gfx1250 async global->LDS path (ASYNCcnt tracked, bypasses VGPRs)
#if defined(__has_builtin)
#if __has_builtin(__builtin_amdgcn_global_load_async_to_lds_b128) && \
    __has_builtin(__builtin_amdgcn_s_wait_asynccnt)
#define HAVE_ASYNC_LDS 1
#endif
#endif
#ifndef HAVE_ASYNC_LDS
#define HAVE_ASYNC_LDS 0
#endif

__device__ __forceinline__ void stage_b128(const float* g, float* l) {
#if HAVE_ASYNC_LDS
    __builtin_amdgcn_global_load_async_to_lds_b128((gv4i*)g, (lv4i*)l, 0, 0);
#else
    *(v4f*)l = *(const v4f*)g;
#endif
}
__device__ __forceinline__ void stage_wait() {
#if HAVE_ASYNC_LDS
    __builtin_amdgcn_s_wait_asynccnt(0);
#endif
}

// ---------------------------------------------------------------------------
// denom[b*N+n] = sum_k adj[b,n,k] + 1
// ---------------------------------------------------------------------------
__global__ __launch_bounds__(256)
void rowsum_kernel(const float* __restrict__ adj, float* __restrict__ denom) {
    __shared__ float red[256];
    const int row = blockIdx.x;
    const float* r = adj + (size_t)row * N_;
    float s = 0.0f;
    #pragma unroll
    for (int i = 0; i < N_ / 256; ++i) s += r[threadIdx.x + i * 256];
    red[threadIdx.x] = s;
    __syncthreads();
    for (int off = 128; off > 0; off >>= 1) {
        if (threadIdx.x < off) red[threadIdx.x] += red[threadIdx.x + off];
        __syncthreads();
    }
    if (threadIdx.x == 0) denom[row] = red[0] + 1.0f;
}

// ---------------------------------------------------------------------------
// Block tile 64 rows x 256 cols, 8 waves, wave = 32 rows (rh) x 64 cols (cq).
// B kept k-pair interleaved in LDS:
//   ldsBp[kp*LDBP + col*2 + (k&1)]  ->  one even-aligned ds_load_b64 per frag.
// ---------------------------------------------------------------------------
#define WMMA_F32(a, b, c) \
    __builtin_amdgcn_wmma_f32_16x16x4_f32(false, (a), false, (b), (short)0, (c), false, false)

// ---------------------------------------------------------------------------
// Y = X @ W^T + bias   (X:[M,256], W:[256,256] row-major [out,in])
// ---------------------------------------------------------------------------
__global__ __launch_bounds__(256)
void xw_kernel(const float* __restrict__ X,
               const float* __restrict__ W,
               const float* __restrict__ bias,
               float* __restrict__ Y) {
    __shared__ float ldsA[64 * LDA];
    __shared__ float ldsBp[16 * LDBP];

    const int t    = threadIdx.x;
    const int wave = t >> 5, lane = t & 31;
    const int l15  = lane & 15, hi = lane >> 4;
    const int rh   = wave & 1;        // 32-row half
    const int cq   = wave >> 1;       // 64-col quarter
    const int m0   = blockIdx.x * 64;

    v8f acc0[4] = {};
    v8f acc1[4] = {};

    for (int k0 = 0; k0 < D_; k0 += 32) {
        __syncthreads();
        // X tile 64x32 (512 x b128; async->LDS)
        #pragma unroll
        for (int i = 0; i < 2; ++i) {
            int f = t + i * 256;
            int row = f >> 3, c4 = (f & 7) * 4;
            stage_b128(X + (size_t)(m0 + row) * D_ + k0 + c4, &ldsA[row * LDA + c4]);
        }
        // W chunk, transposed + k-pair interleaved: ldsBp[kp][m*2+(k&1)] = W[m][k0+k]
        {
            const float* wrow = W + (size_t)t * D_ + k0;   // thread t owns output col m=t
            #pragma unroll
            for (int j = 0; j < 8; ++j) {
                v4f w = *(const v4f*)(wrow + j * 4);
                *(v2f*)(&ldsBp[(2 * j    ) * LDBP + t * 2]) = v2f{w.x, w.y};
                *(v2f*)(&ldsBp[(2 * j + 1) * LDBP + t * 2]) = v2f{w.z, w.w};
            }
        }
        stage_wait();
        __syncthreads();
        #pragma unroll
        for (int k4 = 0; k4 < 8; ++k4) {
            v2f a0 = *(const v2f*)(&ldsA[(rh * 32      + l15) * LDA + k4 * 4 + hi * 2]);
            v2f a1 = *(const v2f*)(&ldsA[(rh * 32 + 16 + l15) * LDA + k4 * 4 + hi * 2]);
            const int kp = k4 * 2 + hi;
            #pragma unroll
            for (int ct = 0; ct < 4; ++ct) {
                int col = cq * 64 + ct * 16 + l15;
                v2f bf = *(const v2f*)(&ldsBp[kp * LDBP + col * 2]);
                acc0[ct] = WMMA_F32(a0, bf, acc0[ct]);
                acc1[ct] = WMMA_F32(a1, bf, acc1[ct]);
            }
        }
    }
    #pragma unroll
    for (int ct = 0; ct < 4; ++ct) {
        int col = cq * 64 + ct * 16 + l15;
        float bb = bias[col];
        #pragma unroll
        for (int v = 0; v < 8; ++v) {
            int r0 = m0 + rh * 32      + v + hi * 8;   // C/D layout: M = vgpr + 8*(lane>=16)
            int r1 = m0 + rh * 32 + 16 + v + hi * 8;
            Y[(size_t)r0 * D_ + col] = acc0[ct][v] + bb;
            Y[(size_t)r1 * D_ + col] = acc1[ct][v] + bb;
        }
    }
}

// ---------------------------------------------------------------------------
// out = relu((adj @ H + G) / denom) [+ resid]     per batch b = blockIdx.y
// ---------------------------------------------------------------------------
__global__ __launch_bounds__(256)
void spmm_kernel(const float* __restrict__ adj,    // [B,N,N]
                 const float* __restrict__ Hm,     // [B,N,D]
                 const float* __restrict__ G,      // [B,N,D]
                 const float* __restrict__ denom,  // [B,N]
                 const float* __restrict__ resid,  // nodes (final layer) or nullptr
                 float* __restrict__ out) {
    __shared__ float ldsA[64 * LDA];
    __shared__ float ldsBp[16 * LDBP];

    const int t    = threadIdx.x;
    const int wave = t >> 5, lane = t & 31;
    const int l15  = lane & 15, hi = lane >> 4;
    const int rh   = wave & 1;
    const int cq   = wave >> 1;
    const int b    = blockIdx.y;
    const int m0   = blockIdx.x * 64;

    const float* adjb = adj + (size_t)b * N_ * N_;
    const float* Hb   = Hm  + (size_t)b * N_ * D_;

    v8f acc0[4] = {};
    v8f acc1[4] = {};

    for (int k0 = 0; k0 < N_; k0 += 32) {
        __syncthreads();
        // adj tile 64x32 (async->LDS)
        #pragma unroll
        for (int i = 0; i < 2; ++i) {
            int f = t + i * 256;
            int row = f >> 3, c4 = (f & 7) * 4;
            stage_b128(adjb + (size_t)(m0 + row) * N_ + k0 + c4, &ldsA[row * LDA + c4]);
        }
        // H tile 32x256, k-pair interleaved: task f -> row-pair r, 4 cols
        #pragma unroll
        for (int i = 0; i < 4; ++i) {
            int f = t + i * 256;
            int r = f >> 6, c = (f & 63) * 4;
            v4f lo = *(const v4f*)(Hb + (size_t)(k0 + 2 * r    ) * D_ + c);
            v4f hh = *(const v4f*)(Hb + (size_t)(k0 + 2 * r + 1) * D_ + c);
            float* bp = &ldsBp[r * LDBP + c * 2];
            *(v2f*)(bp + 0) = v2f{lo.x, hh.x};
            *(v2f*)(bp + 2) = v2f{lo.y, hh.y};
            *(v2f*)(bp + 4) = v2f{lo.z, hh.z};
            *(v2f*)(bp + 6) = v2f{lo.w, hh.w};
        }
        stage_wait();
        __syncthreads();
        #pragma unroll
        for (int k4 = 0; k4 < 8; ++k4) {
            v2f a0 = *(const v2f*)(&ldsA[(rh * 32      + l15) * LDA + k4 * 4 + hi * 2]);
            v2f a1 = *(const v2f*)(&ldsA[(rh * 32 + 16 + l15) * LDA + k4 * 4 + hi * 2]);
            const int kp = k4 * 2 + hi;
            #pragma unroll
            for (int ct = 0; ct < 4; ++ct) {
                int col = cq * 64 + ct * 16 + l15;
                v2f bf = *(const v2f*)(&ldsBp[kp * LDBP + col * 2]);
                acc0[ct] = WMMA_F32(a0, bf, acc0[ct]);
                acc1[ct] = WMMA_F32(a1, bf, acc1[ct]);
            }
        }
    }
    // fused norm + relu (+ residual on final layer)
    const size_t base = (size_t)b * N_ * D_;
    #pragma unroll
    for (int ct = 0; ct < 4; ++ct) {
        int col = cq * 64 + ct * 16 + l15;
        #pragma unroll
        for (int rg2 = 0; rg2 < 2; ++rg2) {
            #pragma unroll
            for (int v = 0; v < 8; ++v) {
                int row = m0 + rh * 32 + rg2 * 16 + v + hi * 8;
                size_t idx = base + (size_t)row * D_ + col;
                float a = (rg2 ? acc1[ct][v] : acc0[ct][v]);
                float r = (a + G[idx]) / denom[b * N_ + row];
                r = fmaxf(r, 0.0f);
                if (resid) r += resid[idx];
                out[idx] = r;
            }
        }
    }
}

// ---------------------------------------------------------------------------
extern "C" void kernel_launch(void* const* d_in, const int* in_sizes, int n_in,
                              void* d_out, int out_size, void* d_ws, size_t ws_size,
                              hipStream_t stream) {
    const float* nodes = (const float*)d_in[0];   // [B,N,D]
    const float* adj   = (const float*)d_in[1];   // [B,N,N]
    const float* W0w   = (const float*)d_in[2];   // [L,D,D]
    const float* W0b   = (const float*)d_in[3];   // [L,D]
    const float* Wrw   = (const float*)d_in[4];   // [L,D,D]
    const float* Wrb   = (const float*)d_in[5];   // [L,D]
    float* out = (float*)d_out;

    const size_t T = (size_t)B_ * N_ * D_;
    float* ws    = (float*)d_ws;
    float* denom = ws;
    float* Hbuf  = ws + (size_t)B_ * N_;
    float* Gbuf  = Hbuf + T;
    float* X1    = Gbuf + T;

    rowsum_kernel<<<B_ * N_, 256, 0, stream>>>(adj, denom);

    dim3 gB((B_ * N_) / 64);
    dim3 gC(N_ / 64, B_);

    // layer 0: x = nodes
    xw_kernel<<<gB, 256, 0, stream>>>(nodes, Wrw,          Wrb,      Hbuf);
    xw_kernel<<<gB, 256, 0, stream>>>(nodes, W0w,          W0b,      Gbuf);
    spmm_kernel<<<gC, 256, 0, stream>>>(adj, Hbuf, Gbuf, denom, nullptr, X1);

    // layer 1: x = X1; fuse final residual (out = nodes + x)
    xw_kernel<<<gB, 256, 0, stream>>>(X1, Wrw + D_ * D_, Wrb + D_, Hbuf);
    xw_kernel<<<gB, 256, 0, stream>>>(X1, W0w + D_ * D_, W0b + D_, Gbuf);
    spmm_kernel<<<gC, 256, 0, stream>>>(adj, Hbuf, Gbuf, denom, nodes, out);
}